// MoEAllGatherTokenDispatcher_9655086482012
// MI455X (gfx1250) — compile-verified
//
#include <hip/hip_runtime.h>
#include <hip/hip_bf16.h>
#include <stdint.h>

// Static problem configuration (matches reference)
#define S 4096
#define B 4
#define H 2048
#define E 64
#define TOPK 2
#define N (S * B)          // 16384 tokens
#define M (N * TOPK)       // 32768 permuted rows
#define CHUNK 256          // tokens per chunk
#define NCHUNK (N / CHUNK) // 64 chunks
#define TDM_ROWS 8         // rows gathered per TDM descriptor (<=16 for 16-bit idx)
#define MPAD (CHUNK + 4)   // padded LDS mask stride (bytes) to avoid bank conflicts

typedef unsigned int u32x4 __attribute__((ext_vector_type(4)));
typedef int          i32x8 __attribute__((ext_vector_type(8)));
typedef int          i32x4 __attribute__((ext_vector_type(4)));

// ---------------------------------------------------------------------------
// K1: per-chunk per-expert routed-token counts.  Mask derived from probs != 0
// (softmax over top-k is strictly positive).  Token-major: each thread reads
// its token's contiguous 256B prob row.
// ---------------------------------------------------------------------------
__global__ void chunk_count_kernel(const float* __restrict__ probs,
                                   int* __restrict__ pc /* [NCHUNK][E] */) {
    __shared__ int hist[E];
    if (threadIdx.x < E) hist[threadIdx.x] = 0;
    __syncthreads();
    const int t = blockIdx.x * CHUNK + threadIdx.x;
    const float* p = probs + (size_t)t * E;
    #pragma unroll
    for (int e = 0; e < E; ++e) {
        if (p[e] != 0.0f) atomicAdd(&hist[e], 1);
    }
    __syncthreads();
    if (threadIdx.x < E) pc[blockIdx.x * E + threadIdx.x] = hist[threadIdx.x];
}

// ---------------------------------------------------------------------------
// K2: expert totals (-> tokens_per_expert), expert-major exclusive scan, and
// per-chunk write bases base[c][e] = offs[e] + sum_{c'<c} pc[c'][e].
// One 64-thread block; 64x64 serial work is trivial.
// ---------------------------------------------------------------------------
__global__ void base_kernel(const int* __restrict__ pc,
                            int* __restrict__ base /* [NCHUNK][E] */,
                            float* __restrict__ tpe_out) {
    __shared__ int cnt[E], offs[E];
    const int e = threadIdx.x;  // blockDim.x == E
    int s = 0;
    for (int c = 0; c < NCHUNK; ++c) s += pc[c * E + e];
    cnt[e] = s;
    tpe_out[e] = (float)s;
    __syncthreads();
    if (e == 0) {
        int a = 0;
        for (int j = 0; j < E; ++j) { offs[j] = a; a += cnt[j]; }
    }
    __syncthreads();
    int run = offs[e];
    for (int c = 0; c < NCHUNK; ++c) {
        base[c * E + e] = run;
        run += pc[c * E + e];
    }
}

// ---------------------------------------------------------------------------
// K3: build token_map (expert-major, stable by token) + per-token weights.
// Phase 1: each thread stashes its token's 64-bit-wide byte mask into LDS
//          (expert-major rows, stride MPAD=260B: coalesced writes, and the
//          per-expert scan phase hits distinct banks), and extracts the two
//          nonzero probs in ascending-expert order (reference's add order).
// Phase 2: thread e walks its expert's 256 flags and appends token indices
//          at base[c][e].
// ---------------------------------------------------------------------------
__global__ void map_kernel(const float* __restrict__ probs,
                           const int* __restrict__ base,
                           int* __restrict__ token_map,
                           float* __restrict__ wts) {
    __shared__ unsigned char lmask[E * MPAD];  // ~16.6 KB
    const int c  = blockIdx.x;
    const int t0 = c * CHUNK;
    const int i  = threadIdx.x;

    const float* p = probs + (size_t)(t0 + i) * E;
    float w0 = 0.0f, w1 = 0.0f;
    int k = 0;
    #pragma unroll
    for (int e = 0; e < E; ++e) {
        const float v = p[e];
        const unsigned char f = (v != 0.0f) ? 1 : 0;
        lmask[e * MPAD + i] = f;
        if (f) {
            if (k == 0) w0 = v; else if (k == 1) w1 = v;
            ++k;
        }
    }
    wts[(size_t)(t0 + i) * 2 + 0] = w0;
    wts[(size_t)(t0 + i) * 2 + 1] = w1;
    __syncthreads();

    if (threadIdx.x < E) {
        const int e = threadIdx.x;
        int pos = base[c * E + e];
        const unsigned* row = (const unsigned*)&lmask[e * MPAD];  // 4-aligned
        for (int j = 0; j < CHUNK / 4; ++j) {
            const unsigned v = row[j];
            if (v) {
                #pragma unroll
                for (int b = 0; b < 4; ++b) {
                    if (v & (0xFFu << (8 * b))) token_map[pos++] = t0 + j * 4 + b;
                }
            }
        }
    }
}

// ---------------------------------------------------------------------------
// K4: output_total[t] = hs[t]*w0 + hs[t]*w1  (streaming float4)
// ---------------------------------------------------------------------------
__global__ void out_kernel(const float* __restrict__ hs,
                           const float* __restrict__ wts,
                           float* __restrict__ out) {
    const int t = blockIdx.x;
    const float w0 = wts[(size_t)t * 2 + 0];
    const float w1 = wts[(size_t)t * 2 + 1];
    const float4* src = (const float4*)(hs + (size_t)t * H);
    float4* dst = (float4*)(out + (size_t)t * H);
    for (int i = threadIdx.x; i < H / 4; i += blockDim.x) {
        const float4 v = src[i];
        float4 r;
        r.x = v.x * w0 + v.x * w1;
        r.y = v.y * w0 + v.y * w1;
        r.z = v.z * w0 + v.z * w1;
        r.w = v.w * w0 + v.w * w1;
        dst[i] = r;
    }
}

// ---------------------------------------------------------------------------
// K5: bulk row gather via the Tensor Data Mover (gather mode).
// One wave per TDM_ROWS permuted rows:
//   tensor_load_to_lds   : gather 8 rows of H floats by 16-bit token index
//   s_wait_tensorcnt 0
//   tensor_store_from_lds: contiguous HxTDM_ROWS tile to permuted[m0..]
// ---------------------------------------------------------------------------
__global__ void permute_tdm_kernel(const float* __restrict__ hs,
                                   const int* __restrict__ token_map,
                                   float* __restrict__ permuted) {
    extern __shared__ float lds[];
    const int m0 = blockIdx.x * TDM_ROWS;

    const unsigned int lds_addr =
        (unsigned int)(unsigned long long)(uintptr_t)&lds[0];  // LDS byte offset
    const unsigned long long src = (unsigned long long)(uintptr_t)hs;
    const unsigned long long dst =
        (unsigned long long)(uintptr_t)(permuted + (size_t)m0 * H);

    // 16-bit gather row indices (token ids < 16384 fit in 16 bits)
    const int* tm = token_map + m0;
    int idx[TDM_ROWS];
    #pragma unroll
    for (int r = 0; r < TDM_ROWS; ++r) idx[r] = tm[r];

    // ---- D# for gather load: group 0 ----
    u32x4 g0l;
    g0l[0] = 0x80000001u;                       // count=1, gather_mode=1, 16-bit idx
    g0l[1] = lds_addr;                          // lds_addr (bytes)
    g0l[2] = (unsigned int)(src & 0xFFFFFFFFull);
    g0l[3] = (unsigned int)((src >> 32) & 0x01FFFFFFull) | 0x80000000u; // type=2

    // ---- group 1 ----
    i32x8 g1l;
    g1l[0] = 0x00020000;                 // workgroup_mask=0, data_size=2 (4 bytes)
    g1l[1] = (int)((unsigned)H << 16);   // tensor_dim0 lo16 @ bits[63:48]
    g1l[2] = (int)((unsigned)N << 16);   // tensor_dim1 lo16 @ bits[95:80]
    g1l[3] = (int)((unsigned)H << 16);   // tile_dim0 @ bits[127:112]
    g1l[4] = TDM_ROWS;                   // tile_dim1 = #valid gather indices
    g1l[5] = H;                          // tensor_dim0_stride (data_size units)
    g1l[6] = 0;
    g1l[7] = 0;

    // ---- groups 2/3: packed 16-bit row indices ----
    i32x4 g2l, g3l;
    g2l[0] = (idx[0] & 0xFFFF) | (idx[1] << 16);
    g2l[1] = (idx[2] & 0xFFFF) | (idx[3] << 16);
    g2l[2] = (idx[4] & 0xFFFF) | (idx[5] << 16);
    g2l[3] = (idx[6] & 0xFFFF) | (idx[7] << 16);
    g3l[0] = 0; g3l[1] = 0; g3l[2] = 0; g3l[3] = 0;

    i32x8 gx;
    gx[0] = 0; gx[1] = 0; gx[2] = 0; gx[3] = 0;
    gx[4] = 0; gx[5] = 0; gx[6] = 0; gx[7] = 0;

    __builtin_amdgcn_tensor_load_to_lds(g0l, g1l, g2l, g3l, gx, 0);
    __builtin_amdgcn_s_wait_tensorcnt(0);

    // ---- D# for contiguous store: 2D tile H x TDM_ROWS ----
    u32x4 g0s;
    g0s[0] = 0x00000001u;                // count=1, normal mode
    g0s[1] = lds_addr;
    g0s[2] = (unsigned int)(dst & 0xFFFFFFFFull);
    g0s[3] = (unsigned int)((dst >> 32) & 0x01FFFFFFull) | 0x80000000u; // type=2

    i32x8 g1s;
    g1s[0] = 0x00020000;                      // data_size = 4 bytes
    g1s[1] = (int)((unsigned)H << 16);        // tensor_dim0 = H
    g1s[2] = (int)((unsigned)TDM_ROWS << 16); // tensor_dim1 = TDM_ROWS
    g1s[3] = (int)((unsigned)H << 16);        // tile_dim0 = H
    g1s[4] = TDM_ROWS;                        // tile_dim1 = TDM_ROWS
    g1s[5] = H;                               // tensor_dim0_stride
    g1s[6] = 0;
    g1s[7] = 0;

    i32x4 gz;
    gz[0] = 0; gz[1] = 0; gz[2] = 0; gz[3] = 0;

    __builtin_amdgcn_tensor_store_from_lds(g0s, g1s, gz, gz, gx, 0);
    __builtin_amdgcn_s_wait_tensorcnt(0);
}

// ---------------------------------------------------------------------------
// Launch
// ---------------------------------------------------------------------------
extern "C" void kernel_launch(void* const* d_in, const int* in_sizes, int n_in,
                              void* d_out, int out_size, void* d_ws, size_t ws_size,
                              hipStream_t stream) {
    const float* hs    = (const float*)d_in[0];   // [N, H]
    const float* probs = (const float*)d_in[1];   // [N, E]
    // d_in[2] (routing_map) intentionally unused: mask == (probs != 0)

    float* out_total = (float*)d_out;                 // [N*H]
    float* out_tpe   = out_total + (size_t)N * H;     // [E]
    float* out_perm  = out_tpe + E;                   // [M*H]

    char* ws = (char*)d_ws;
    int*   pc        = (int*)(ws);                                  // NCHUNK*E ints
    int*   base      = (int*)(ws + sizeof(int) * NCHUNK * E);       // NCHUNK*E ints
    int*   token_map = (int*)(ws + 2 * sizeof(int) * NCHUNK * E);   // M ints
    float* wts       = (float*)(ws + 2 * sizeof(int) * NCHUNK * E
                                   + sizeof(int) * (size_t)M);      // N*2 floats

    chunk_count_kernel<<<NCHUNK, CHUNK, 0, stream>>>(probs, pc);
    base_kernel<<<1, E, 0, stream>>>(pc, base, out_tpe);
    map_kernel<<<NCHUNK, CHUNK, 0, stream>>>(probs, base, token_map, wts);
    out_kernel<<<N, 256, 0, stream>>>(hs, wts, out_total);
    permute_tdm_kernel<<<M / TDM_ROWS, 32, TDM_ROWS * H * sizeof(float), stream>>>(
        hs, token_map, out_perm);
}